// TransformFeatureQuaternion_82480551952586
// MI455X (gfx1250) — compile-verified
//
#include <hip/hip_runtime.h>

typedef __attribute__((ext_vector_type(2))) float v2f;
typedef __attribute__((ext_vector_type(4))) float v4f;
typedef __attribute__((ext_vector_type(8))) float v8f;

#define FEAT_DIM 480
#define L0_DIM   128
#define L1_OFF   128
#define L2_OFF   320

// ---------------------------------------------------------------------------
// Wigner-D precompute: one lane per t. Builds D1 (3x3) and D2 (5x5) in d_ws.
// ---------------------------------------------------------------------------
template <int D>
__device__ inline void zrot(float ang, float* M) {
  const int l = (D - 1) / 2;
#pragma unroll
  for (int x = 0; x < D * D; ++x) M[x] = 0.0f;
#pragma unroll
  for (int i = 0; i < D; ++i) {           // anti-diagonal: sin(freq*ang)
    float f = (float)(l - i);
    M[i * D + (D - 1 - i)] = sinf(f * ang);
  }
#pragma unroll
  for (int i = 0; i < D; ++i) {           // diagonal: cos(freq*ang) (center -> 1)
    float f = (float)(l - i);
    M[i * D + i] = cosf(f * ang);
  }
}

template <int D>
__device__ inline void matmul(const float* A, const float* B, float* C) {
#pragma unroll
  for (int a = 0; a < D; ++a)
#pragma unroll
    for (int b = 0; b < D; ++b) {
      float s = 0.0f;
#pragma unroll
      for (int k = 0; k < D; ++k) s += A[a * D + k] * B[k * D + b];
      C[a * D + b] = s;
    }
}

__global__ void wigner_kernel(const float* __restrict__ q,
                              const float* __restrict__ J1,
                              const float* __restrict__ J2,
                              float* __restrict__ D1,
                              float* __restrict__ D2, int T) {
  int t = blockIdx.x * blockDim.x + threadIdx.x;
  if (t >= T) return;
  float r = q[4 * t + 0], i = q[4 * t + 1], j = q[4 * t + 2], k = q[4 * t + 3];
  float inv = rsqrtf(r * r + i * i + j * j + k * k);
  r *= inv; i *= inv; j *= inv; k *= inv;
  float two_s = 2.0f / (r * r + i * i + j * j + k * k);
  float M01 = two_s * (i * j - k * r);
  float M21 = two_s * (j * k + i * r);
  float M11 = 1.0f - two_s * (i * i + k * k);
  float M10 = two_s * (i * j + k * r);
  float M12 = two_s * (j * k - i * r);
  float alpha = atan2f(M01, M21);
  float beta  = acosf(fminf(1.0f, fmaxf(-1.0f, M11)));
  float gamma = atan2f(M10, -M12);
  {
    float Za[9], Zb[9], Zg[9], T1[9], T2[9], Jl[9];
#pragma unroll
    for (int x = 0; x < 9; ++x) Jl[x] = J1[x];
    zrot<3>(alpha, Za); zrot<3>(beta, Zb); zrot<3>(gamma, Zg);
    matmul<3>(Za, Jl, T1); matmul<3>(T1, Zb, T2);
    matmul<3>(T2, Jl, T1); matmul<3>(T1, Zg, T2);
#pragma unroll
    for (int x = 0; x < 9; ++x) D1[t * 9 + x] = T2[x];
  }
  {
    float Za[25], Zb[25], Zg[25], T1[25], T2[25], Jl[25];
#pragma unroll
    for (int x = 0; x < 25; ++x) Jl[x] = J2[x];
    zrot<5>(alpha, Za); zrot<5>(beta, Zb); zrot<5>(gamma, Zg);
    matmul<5>(Za, Jl, T1); matmul<5>(T1, Zb, T2);
    matmul<5>(T2, Jl, T1); matmul<5>(T1, Zg, T2);
#pragma unroll
    for (int x = 0; x < 25; ++x) D2[t * 25 + x] = T2[x];
  }
}

// ---------------------------------------------------------------------------
// l=0: broadcast copy (float4, non-temporal stores)
// ---------------------------------------------------------------------------
__global__ void l0_copy_kernel(const float* __restrict__ feat,
                               float* __restrict__ out, int N, int T) {
  long long idx = (long long)blockIdx.x * blockDim.x + threadIdx.x;
  long long total = (long long)T * N * (L0_DIM / 4);
  if (idx >= total) return;
  int c = (int)(idx & (L0_DIM / 4 - 1));   // 32 float4 chunks per row
  long long rem = idx >> 5;
  int p = (int)(rem % N);
  int t = (int)(rem / N);
  const v4f* src = (const v4f*)(feat + (long long)p * FEAT_DIM) + c;
  v4f v = *src;
  v4f* dst = (v4f*)(out + ((long long)t * N + p) * FEAT_DIM) + c;
  __builtin_nontemporal_store(v, dst);
}

// ---------------------------------------------------------------------------
// l=1 rotation via V_WMMA_F32_16X16X4_F32, two t-values packed per wave:
//   A(16x4): rows 0..2 = D1[t0], rows 8..10 = D1[t1], K col 3 = 0
//   B(4x16): B[b][col] = fr[p, nbase+col, b]  (b=3 pad zeroed by cndmask)
//   C(16x16): lanes 0-15 hold t0 results (a in c[0..2]),
//             lanes 16-31 hold t1 results -> all 32 lanes do b96 NT stores.
// One wave per (t-pair, p); 4 tiles of 16 rows; A reused across tiles.
// ---------------------------------------------------------------------------
__global__ void l1_rot_kernel(const float* __restrict__ feat,
                              const float* __restrict__ D1,
                              float* __restrict__ out, int N, int T) {
  int lane = threadIdx.x & 31;
  int half = lane >> 4, col = lane & 15;
  long long wid = ((long long)blockIdx.x * blockDim.x + threadIdx.x) >> 5;
  int Tp = (T + 1) >> 1;
  long long nwaves = (long long)Tp * N;
  if (wid >= nwaves) return;          // wave-uniform: EXEC stays all-1s
  int p  = (int)(wid % N);
  int tp = (int)(wid / N);
  int t0 = tp * 2;
  int t1 = (t0 + 1 < T) ? t0 + 1 : T - 1;
  int k0 = half * 2;

  // A fragment: row = col (0..2 -> t0, 8..10 -> t1), k = k0 + v
  v2f A; A.x = 0.0f; A.y = 0.0f;
  {
    int r = col & 7;
    const float* Dsel = D1 + ((col < 8) ? t0 : t1) * 9;
    if (r < 3) {
      A.x = Dsel[r * 3 + k0];                       // k = 0 or 2
      A.y = (k0 + 1 < 3) ? Dsel[r * 3 + k0 + 1] : 0.0f;
    }
  }

  const float* frow = feat + (long long)p * FEAT_DIM + L1_OFF;
  int tsel = (half == 0) ? t0 : t1;   // which t this lane stores
  float* orow = out + ((long long)tsel * N + p) * FEAT_DIM + L1_OFF;
#pragma unroll
  for (int tile = 0; tile < 4; ++tile) {
    const float* fb = frow + (tile * 16 + col) * 3;
    // Unconditional loads (fb[3] is in-bounds inside the 480-float row and
    // multiplies an exactly-zero A column); zero pad lane via select, no branch.
    float bx = fb[k0];
    float by = fb[k0 + 1];
    v2f B;
    B.x = bx;
    B.y = (half == 0) ? by : 0.0f;    // b=1 valid, b=3 -> 0
    v8f c = {};
    c = __builtin_amdgcn_wmma_f32_16x16x4_f32(false, A, false, B,
                                              (short)0, c, false, false);
    float* op = orow + (tile * 16 + col) * 3;
    __builtin_nontemporal_store(c[0], op + 0);
    __builtin_nontemporal_store(c[1], op + 1);
    __builtin_nontemporal_store(c[2], op + 2);
  }
}

// ---------------------------------------------------------------------------
// l=2 rotation: K=5 -> two chained WMMAs (b=0..3, then b=4), two t per wave.
// ---------------------------------------------------------------------------
__global__ void l2_rot_kernel(const float* __restrict__ feat,
                              const float* __restrict__ D2,
                              float* __restrict__ out, int N, int T) {
  int lane = threadIdx.x & 31;
  int half = lane >> 4, col = lane & 15;
  long long wid = ((long long)blockIdx.x * blockDim.x + threadIdx.x) >> 5;
  int Tp = (T + 1) >> 1;
  long long nwaves = (long long)Tp * N;
  if (wid >= nwaves) return;
  int p  = (int)(wid % N);
  int tp = (int)(wid / N);
  int t0 = tp * 2;
  int t1 = (t0 + 1 < T) ? t0 + 1 : T - 1;
  int k0 = half * 2;

  // A fragments: rows 0..4 = D2[t0], rows 8..12 = D2[t1]
  v2f A0; A0.x = 0.0f; A0.y = 0.0f;
  v2f A1; A1.x = 0.0f; A1.y = 0.0f;
  {
    int r = col & 7;
    const float* Dsel = D2 + ((col < 8) ? t0 : t1) * 25;
    if (r < 5) {
      A0.x = Dsel[r * 5 + k0];          // k = 0,1 (half 0) / 2,3 (half 1)
      A0.y = Dsel[r * 5 + k0 + 1];
      if (half == 0) A1.x = Dsel[r * 5 + 4];   // chunk 2: only k=4 live
    }
  }

  const float* frow = feat + (long long)p * FEAT_DIM + L2_OFF;
  int tsel = (half == 0) ? t0 : t1;
  float* orow = out + ((long long)tsel * N + p) * FEAT_DIM + L2_OFF;
#pragma unroll
  for (int tile = 0; tile < 2; ++tile) {
    const float* fb = frow + (tile * 16 + col) * 5;
    float b0x = fb[k0];
    float b1x = fb[k0 + 1];
    float b4  = fb[4];                  // same address both halves; in-bounds
    v2f B0, B1;
    B0.x = b0x;
    B0.y = b1x;                         // b = 0..3 all valid
    B1.x = (half == 0) ? b4 : 0.0f;     // chunk-2 row 0 = b=4; rows 2,3 -> 0
    B1.y = 0.0f;                        // rows 1,(3) -> 0
    v8f c = {};
    c = __builtin_amdgcn_wmma_f32_16x16x4_f32(false, A0, false, B0,
                                              (short)0, c, false, false);
    c = __builtin_amdgcn_wmma_f32_16x16x4_f32(false, A1, false, B1,
                                              (short)0, c, false, false);
    float* op = orow + (tile * 16 + col) * 5;
    __builtin_nontemporal_store(c[0], op + 0);
    __builtin_nontemporal_store(c[1], op + 1);
    __builtin_nontemporal_store(c[2], op + 2);
    __builtin_nontemporal_store(c[3], op + 3);
    __builtin_nontemporal_store(c[4], op + 4);
  }
}

// ---------------------------------------------------------------------------
extern "C" void kernel_launch(void* const* d_in, const int* in_sizes, int n_in,
                              void* d_out, int out_size, void* d_ws, size_t ws_size,
                              hipStream_t stream) {
  const float* feat = (const float*)d_in[0];
  const float* q    = (const float*)d_in[1];
  // d_in[2] = J0 (1x1) unused: l=0 block is a pure broadcast in the reference.
  const float* J1   = (const float*)d_in[3];
  const float* J2   = (const float*)d_in[4];
  float* out = (float*)d_out;

  const int T = in_sizes[1] / 4;
  const int N = in_sizes[0] / FEAT_DIM;

  float* D1 = (float*)d_ws;        // T*9 floats
  float* D2 = D1 + (size_t)T * 9;  // T*25 floats

  // 1) Wigner-D matrices (tiny)
  {
    int threads = 32;
    int blocks = (T + threads - 1) / threads;
    wigner_kernel<<<blocks, threads, 0, stream>>>(q, J1, J2, D1, D2, T);
  }
  // 2) l=0 broadcast copy
  {
    long long total = (long long)T * N * (L0_DIM / 4);
    int threads = 256;
    long long blocks = (total + threads - 1) / threads;
    l0_copy_kernel<<<(unsigned int)blocks, threads, 0, stream>>>(feat, out, N, T);
  }
  // 3) l=1 rotation (WMMA), one wave per (t-pair, p)
  {
    int Tp = (T + 1) / 2;
    long long waves = (long long)Tp * N;
    int threads = 256;  // 8 waves/block
    long long blocks = (waves * 32 + threads - 1) / threads;
    l1_rot_kernel<<<(unsigned int)blocks, threads, 0, stream>>>(feat, D1, out, N, T);
  }
  // 4) l=2 rotation (WMMA), one wave per (t-pair, p)
  {
    int Tp = (T + 1) / 2;
    long long waves = (long long)Tp * N;
    int threads = 256;
    long long blocks = (waves * 32 + threads - 1) / threads;
    l2_rot_kernel<<<(unsigned int)blocks, threads, 0, stream>>>(feat, D2, out, N, T);
  }
}